// MultiHeadAttention_31112743092867
// MI455X (gfx1250) — compile-verified
//
#include <hip/hip_runtime.h>
#include <math.h>

typedef __attribute__((ext_vector_type(2))) float v2f;
typedef __attribute__((ext_vector_type(8))) float v8f;

#define BATCH 2
#define SK 32768
#define DIM 1024
#define NH 16
#define DH 64
#define SCALE 0.125f          // 1/sqrt(64)
#define NJC 32                // split-K chunks for probs@V
#define JCH (SK / NJC)        // 1024 rows per chunk

// -------- workspace layout (float offsets) --------
// qp  [2][1024]
// UF  [2][256 ksteps][32 lanes][2]   (B-fragment layout of U = Wk_slice @ qp)
// sb  [2][16]                         (qp . bk_slice)
// SC  [2][16][32768]                  scores, softmaxed in place -> probs
// TP  [2][32][16][1024]               split-K partials of t = probs^T @ v
// T   [2][16][1024]                   reduced t
#define QP_OFF 0
#define UF_OFF 2048
#define SB_OFF (UF_OFF + BATCH*NH*DIM)          // 34816
#define SC_OFF (SB_OFF + 32)                    // 34848
#define TP_OFF (SC_OFF + BATCH*NH*SK)           // + 1048576
#define T_OFF  (TP_OFF + BATCH*NJC*NH*DIM)      // + 1048576
// total = T_OFF + 32768 floats  ~= 8.26 MiB of d_ws

// ---------------- 1) qp = q @ Wq + bq ----------------
__global__ void qproj_kernel(const float* __restrict__ q, const float* __restrict__ Wq,
                             const float* __restrict__ bq, float* __restrict__ ws) {
  int b = blockIdx.x, tid = threadIdx.x;
  const float* qr = q + (size_t)b * DIM;
  for (int i = 0; i < 4; ++i) {
    int c = tid + 256 * i;
    float acc = bq[c];
    for (int e = 0; e < DIM; ++e) acc += qr[e] * Wq[(size_t)e * DIM + c];
    ws[QP_OFF + b * DIM + c] = acc;
  }
}

// ------- 2) U fragments (WMMA B layout) + score bias -------
__global__ void ufrag_kernel(const float* __restrict__ Wk, const float* __restrict__ bk,
                             float* __restrict__ ws) {
  __shared__ float qps[DH];
  __shared__ float red[256];
  int b = blockIdx.x / NH, h = blockIdx.x % NH, tid = threadIdx.x;
  if (tid < DH) qps[tid] = ws[QP_OFF + b * DIM + h * DH + tid];
  __syncthreads();
  for (int i = 0; i < 4; ++i) {
    int e = tid + 256 * i;
    float u = 0.f;
    const float* wr = Wk + (size_t)e * DIM + h * DH;
    for (int d = 0; d < DH; ++d) u += wr[d] * qps[d];
    // B-fragment layout for V_WMMA_F32_16X16X4_F32:
    // lane = n + 16*(k>=2), vreg r = k&1, here n=h, k=e%4
    int estep = e >> 2, sub = e & 3;
    int lane = h + 16 * (sub >> 1), r = sub & 1;
    ws[UF_OFF + ((size_t)(b * 256 + estep) * 32 + lane) * 2 + r] = u;
  }
  float lb = (tid < DH) ? bk[h * DH + tid] * qps[tid] : 0.f;
  red[tid] = lb; __syncthreads();
  for (int s = 128; s > 0; s >>= 1) { if (tid < s) red[tid] += red[tid + s]; __syncthreads(); }
  if (tid == 0) ws[SB_OFF + b * NH + h] = red[0];
}

// ------- 3) scores = (k @ U^T + sb) * scale  (streams k once, WMMA f32) -------
__global__ void __launch_bounds__(256)
scores_kernel(const float* __restrict__ k, float* __restrict__ ws, float* __restrict__ out) {
  int tid = threadIdx.x, lane = tid & 31;
  int w  = blockIdx.x * 8 + (tid >> 5);   // global wave id, 4096 total
  int b  = w >> 11;                       // / 2048 tiles per batch
  int jt = w & 2047;
  int j0 = jt << 4;
  int ln15 = lane & 15, half = lane >> 4;
  const float* krow = k + ((size_t)(b * SK + j0 + ln15)) * DIM + 2 * half; // A-frag base
  const float* uf   = ws + UF_OFF + ((size_t)b * 256 * 32 + lane) * 2;    // B-frag base
  v8f acc = {};
  for (int estep = 0; estep < 256; ++estep) {
    v2f a  = *(const v2f*)(krow + estep * 4);
    v2f bb = *(const v2f*)(uf + (size_t)estep * 64);
    __builtin_prefetch(krow + estep * 4 + 256, 0, 3);   // global_prefetch_b8
    acc = __builtin_amdgcn_wmma_f32_16x16x4_f32(false, a, false, bb, (short)0, acc,
                                                false, false);
  }
  float sb = ws[SB_OFF + b * NH + ln15];
  // D layout: vreg r, lanes0-15 -> (M=r, N=lane); lanes16-31 -> (M=r+8, N=lane-16)
  for (int r = 0; r < 8; ++r) {
    int j = j0 + r + 8 * half;
    float val = (acc[r] + sb) * SCALE;
    ws[SC_OFF + ((size_t)(b * NH + ln15)) * SK + j] = val;
    if (b == 0) out[2048 + (size_t)ln15 * SK + j] = val;   // attn output (raw scaled scores)
  }
}

// ---------------- 4) softmax in place over j ----------------
__global__ void softmax_kernel(float* __restrict__ ws) {
  __shared__ float red[256];
  int bh = blockIdx.x, tid = threadIdx.x;
  float* S = ws + SC_OFF + (size_t)bh * SK;
  float m = -1e30f;
  for (int i = tid; i < SK; i += 256) m = fmaxf(m, S[i]);
  red[tid] = m; __syncthreads();
  for (int s = 128; s > 0; s >>= 1) { if (tid < s) red[tid] = fmaxf(red[tid], red[tid + s]); __syncthreads(); }
  m = red[0]; __syncthreads();
  float sum = 0.f;
  for (int i = tid; i < SK; i += 256) sum += __expf(S[i] - m);
  red[tid] = sum; __syncthreads();
  for (int s = 128; s > 0; s >>= 1) { if (tid < s) red[tid] += red[tid + s]; __syncthreads(); }
  float inv = 1.f / red[0];
  for (int i = tid; i < SK; i += 256) S[i] = __expf(S[i] - m) * inv;
}

// ------- 5) t_partial = probs^T @ v  (streams v once, WMMA f32, split-K) -------
__global__ void __launch_bounds__(256)
attnv_kernel(const float* __restrict__ v, float* __restrict__ ws) {
  __shared__ float sp[NH][JCH + 2];   // padded stride: avoid 64-bank conflicts
  int tid = threadIdx.x;
  int blk = blockIdx.x;               // 256 blocks = 2b * 32jc * 4eblk
  int b    = blk >> 7;
  int rem  = blk & 127;
  int jc   = rem >> 2;
  int eblk = rem & 3;
  int lane = tid & 31, wave = tid >> 5;
  int e0 = eblk * 256 + wave * 32;    // this wave: columns e0..e0+31
  int j0 = jc * JCH;
  const float* P = ws + SC_OFF;
  for (int i = 0; i < (NH * JCH) / 256; ++i) {   // stage probs chunk in LDS
    int f = tid + 256 * i;
    int hh = f >> 10, jj = f & (JCH - 1);
    sp[hh][jj] = P[((size_t)(b * NH + hh)) * SK + j0 + jj];
  }
  __syncthreads();
  int ln15 = lane & 15, half = lane >> 4;
  const float* vb = v + ((size_t)b * SK + j0) * DIM;
  v8f c0 = {}, c1 = {};
  for (int kk = 0; kk < JCH; kk += 4) {
    v2f a = *(const v2f*)&sp[ln15][kk + 2 * half];          // A-frag (probs)
    const float* vr = vb + (size_t)(kk + 2 * half) * DIM;
    v2f b0, b1;
    b0.x = vr[e0 + ln15];        b0.y = vr[DIM + e0 + ln15];
    b1.x = vr[e0 + 16 + ln15];   b1.y = vr[DIM + e0 + 16 + ln15];
    c0 = __builtin_amdgcn_wmma_f32_16x16x4_f32(false, a, false, b0, (short)0, c0, false, false);
    c1 = __builtin_amdgcn_wmma_f32_16x16x4_f32(false, a, false, b1, (short)0, c1, false, false);
  }
  float* tp = ws + TP_OFF + ((size_t)(b * NJC + jc) * NH) * DIM;
  for (int r = 0; r < 8; ++r) {
    int mm = r + 8 * half;        // head index
    tp[(size_t)mm * DIM + e0 + ln15]      = c0[r];
    tp[(size_t)mm * DIM + e0 + 16 + ln15] = c1[r];
  }
}

// ---------------- 6) reduce split-K partials ----------------
__global__ void reduce_t_kernel(float* __restrict__ ws) {
  int f = blockIdx.x * 256 + threadIdx.x;   // 0..32767 = (b,h,e)
  int b = f >> 14, rest = f & 16383;
  const float* tp = ws + TP_OFF + (size_t)b * NJC * NH * DIM + rest;
  float s = 0.f;
  for (int jc = 0; jc < NJC; ++jc) s += tp[(size_t)jc * NH * DIM];
  ws[T_OFF + f] = s;
}

// ------- 7) out = LN( (t@Wv_slice + bv) @ Wo + bo ) -> feat -------
__global__ void final_kernel(const float* __restrict__ Wv, const float* __restrict__ bv,
                             const float* __restrict__ Wo, const float* __restrict__ bo,
                             const float* __restrict__ lnw, const float* __restrict__ lnb,
                             const float* __restrict__ ws, float* __restrict__ out) {
  __shared__ float som[DIM];
  __shared__ float sy[DIM];
  __shared__ float red[256];
  int b = blockIdx.x, tid = threadIdx.x;
  const float* t = ws + T_OFF + (size_t)b * NH * DIM;
  for (int i = 0; i < 4; ++i) {        // merged-head projection through Wv slice
    int c = tid + 256 * i;
    int h = c >> 6;
    float acc = bv[c];                 // sum(probs)=1 makes bv exact
    const float* th = t + (size_t)h * DIM;
    for (int e = 0; e < DIM; ++e) acc += th[e] * Wv[(size_t)e * DIM + c];
    som[c] = acc;
  }
  __syncthreads();
  for (int i = 0; i < 4; ++i) {        // output projection
    int c2 = tid + 256 * i;
    float acc = bo[c2];
    for (int c = 0; c < DIM; ++c) acc += som[c] * Wo[(size_t)c * DIM + c2];
    sy[c2] = acc;
  }
  __syncthreads();
  float ls = 0.f;
  for (int i = 0; i < 4; ++i) ls += sy[tid + 256 * i];
  red[tid] = ls; __syncthreads();
  for (int s = 128; s > 0; s >>= 1) { if (tid < s) red[tid] += red[tid + s]; __syncthreads(); }
  float mu = red[0] / (float)DIM;
  __syncthreads();
  float lv = 0.f;
  for (int i = 0; i < 4; ++i) { float d = sy[tid + 256 * i] - mu; lv += d * d; }
  red[tid] = lv; __syncthreads();
  for (int s = 128; s > 0; s >>= 1) { if (tid < s) red[tid] += red[tid + s]; __syncthreads(); }
  float inv = rsqrtf(red[0] / (float)DIM + 1e-6f);
  for (int i = 0; i < 4; ++i) {
    int c2 = tid + 256 * i;
    out[b * DIM + c2] = (sy[c2] - mu) * inv * lnw[c2] + lnb[c2];
  }
}

extern "C" void kernel_launch(void* const* d_in, const int* in_sizes, int n_in,
                              void* d_out, int out_size, void* d_ws, size_t ws_size,
                              hipStream_t stream) {
  const float* q   = (const float*)d_in[0];
  const float* k   = (const float*)d_in[1];
  const float* v   = (const float*)d_in[2];
  const float* Wq  = (const float*)d_in[3];
  const float* bq  = (const float*)d_in[4];
  const float* Wk  = (const float*)d_in[5];
  const float* bk  = (const float*)d_in[6];
  const float* Wv  = (const float*)d_in[7];
  const float* bv  = (const float*)d_in[8];
  const float* Wo  = (const float*)d_in[9];
  const float* bo  = (const float*)d_in[10];
  const float* lnw = (const float*)d_in[11];
  const float* lnb = (const float*)d_in[12];
  float* out = (float*)d_out;
  float* ws  = (float*)d_ws;

  qproj_kernel   <<<BATCH,        256, 0, stream>>>(q, Wq, bq, ws);
  ufrag_kernel   <<<BATCH * NH,   256, 0, stream>>>(Wk, bk, ws);
  scores_kernel  <<<BATCH * SK / 16 / 8, 256, 0, stream>>>(k, ws, out);   // 512 blocks
  softmax_kernel <<<BATCH * NH,   256, 0, stream>>>(ws);
  attnv_kernel   <<<BATCH * NJC * 4, 256, 0, stream>>>(v, ws);            // 256 blocks
  reduce_t_kernel<<<BATCH * NH * DIM / 256, 256, 0, stream>>>(ws);        // 128 blocks
  final_kernel   <<<BATCH,        256, 0, stream>>>(Wv, bv, Wo, bo, lnw, lnb, ws, out);
}